// SafetyLayer_28484223107561
// MI455X (gfx1250) — compile-verified
//
#include <hip/hip_runtime.h>
#include <hip/hip_bf16.h>
#include <stdint.h>

#define FINAL_SOC 1.0f
#define ALPHA_C   10.0f
#define ALPHA_D   10.0f
#define B_CAP     40.0f
#define ETA_C     0.95f
#define ETA_D     0.95f
#define BIG       1000.0f
#define D_CARS    256
#define BATCH     8192

// ---------------------------------------------------------------------------
// Kernel 1: compute y_min / y_max per car (256 cars). One block, 256 threads.
// Mirrors _bounds() + _y_interval() from the reference exactly.
// Output layout in workspace: ws[0..255] = y_min, ws[256..511] = y_max.
// ---------------------------------------------------------------------------
__global__ void __launch_bounds__(D_CARS)
safety_bounds_kernel(const float* __restrict__ obs, float* __restrict__ ws) {
    const int d = threadIdx.x;   // car index, 0..255

    // obs packed as [soc_t, t_rem, soc_dis, t_dis] per car -> one b128 load
    float4 st = reinterpret_cast<const float4*>(obs)[d];
    const float soc_t   = st.x;
    const float t_rem   = st.y;
    const float soc_dis = st.z;
    const float t_dis   = st.w;

    const bool active  = t_rem > 0.0f;
    const bool can_dis = active && (t_dis > 0.0f);

    const float a_lo = 0.0f;
    const float a_hi = active ? fminf(ALPHA_C / B_CAP, BIG * t_rem) : 0.0f;
    const float b_lo = can_dis
        ? fmaxf(fmaxf(-(ALPHA_D / B_CAP), -ETA_D * soc_dis),
                fmaxf(-BIG * t_rem, -BIG * t_dis))
        : 0.0f;
    const float b_hi = 0.0f;
    const float s_lo = active
        ? (fmaxf(0.0f, FINAL_SOC - ((t_rem - 1.0f) * ALPHA_C) * ETA_C / B_CAP) - soc_t)
        : -1000000000.0f;
    const float s_hi = active ? (FINAL_SOC - soc_t) : 1000000000.0f;

    const float c1  = ETA_C;
    const float c2  = 1.0f / ETA_D;
    const float eps = 1e-5f;

    float ymin =  INFINITY;
    float ymax = -INFINITY;

    auto consider = [&](float A, float B) {
        const float S = c1 * A + c2 * B;
        const bool feas =
            (A >= a_lo - eps) && (A <= a_hi + eps) &&
            (B >= b_lo - eps) && (B <= b_hi + eps) &&
            (S >= s_lo - eps) && (S <= s_hi + eps);
        if (feas) {
            const float Y = A + B;
            ymin = fminf(ymin, Y);
            ymax = fmaxf(ymax, Y);
        }
    };

    // 4 corners of the (a,b) box
    consider(a_lo, b_lo);
    consider(a_lo, b_hi);
    consider(a_hi, b_lo);
    consider(a_hi, b_hi);
    // A fixed, S fixed -> solve B
    consider(a_lo, (s_lo - c1 * a_lo) / c2);
    consider(a_hi, (s_lo - c1 * a_hi) / c2);
    consider(a_lo, (s_hi - c1 * a_lo) / c2);
    consider(a_hi, (s_hi - c1 * a_hi) / c2);
    // B fixed, S fixed -> solve A
    consider((s_lo - c2 * b_lo) / c1, b_lo);
    consider((s_lo - c2 * b_hi) / c1, b_hi);
    consider((s_hi - c2 * b_lo) / c1, b_lo);
    consider((s_hi - c2 * b_hi) / c1, b_hi);

    ws[d]          = ymin;
    ws[D_CARS + d] = ymax;
}

// ---------------------------------------------------------------------------
// Kernel 2: stream-clip x[8192,256] against the per-column bounds.
// HBM-bound: ~16.8 MB traffic -> ~0.72 us at 23.3 TB/s.
// Bounds table (2 KB) is staged into LDS with the CDNA5 async global->LDS DMA
// (global_load_async_to_lds_b128, ASYNCcnt), then every thread streams float4s.
// ---------------------------------------------------------------------------
__global__ void __launch_bounds__(256)
safety_clip_kernel(const float* __restrict__ x,
                   const float* __restrict__ bounds,
                   float* __restrict__ out, int n4) {
    __shared__ __align__(16) float sb[2 * D_CARS];   // [0..255]=ymin, [256..511]=ymax

    // 128 lanes x 16B = 2 KB: async DMA global -> LDS, no VGPR round-trip.
    if (threadIdx.x < 128) {
        unsigned long long gaddr =
            (unsigned long long)(const void*)(bounds + threadIdx.x * 4);
        unsigned lds_off =
            (unsigned)(uintptr_t)(const void*)&sb[threadIdx.x * 4];
        asm volatile("global_load_async_to_lds_b128 %0, %1, off"
                     :: "v"(lds_off), "v"(gaddr) : "memory");
        asm volatile("s_wait_asynccnt 0x0" ::: "memory");
    }
    __syncthreads();

    const float4* __restrict__ x4 = reinterpret_cast<const float4*>(x);
    float4* __restrict__ o4       = reinterpret_cast<float4*>(out);

    const int tid    = blockIdx.x * blockDim.x + threadIdx.x;
    const int stride = gridDim.x * blockDim.x;

    for (int e = tid; e < n4; e += stride) {
        if (e + stride < n4) {
            __builtin_prefetch(&x4[e + stride], 0, 0);  // global_prefetch_b8
        }
        const float4 xv = x4[e];
        const int c = (e & (D_CARS / 4 - 1)) * 4;       // column group -> column
        const float4 mn = *reinterpret_cast<const float4*>(&sb[c]);
        const float4 mx = *reinterpret_cast<const float4*>(&sb[D_CARS + c]);
        float4 r;
        r.x = fminf(mx.x, fmaxf(mn.x, xv.x));           // == jnp.clip ordering
        r.y = fminf(mx.y, fmaxf(mn.y, xv.y));
        r.z = fminf(mx.z, fmaxf(mn.z, xv.z));
        r.w = fminf(mx.w, fmaxf(mn.w, xv.w));
        o4[e] = r;
    }
}

// ---------------------------------------------------------------------------
extern "C" void kernel_launch(void* const* d_in, const int* in_sizes, int n_in,
                              void* d_out, int out_size, void* d_ws, size_t ws_size,
                              hipStream_t stream) {
    const float* x   = (const float*)d_in[0];   // (8192, 256) f32
    const float* obs = (const float*)d_in[1];   // (1, 1024)   f32
    float* out = (float*)d_out;                 // (8192, 256) f32
    float* ws  = (float*)d_ws;                  // 512 floats of scratch

    // Phase 1: per-car interval (one tiny block)
    safety_bounds_kernel<<<1, D_CARS, 0, stream>>>(obs, ws);

    // Phase 2: bandwidth-bound clip. 1024 blocks x 256 thr (8 wave32/block),
    // grid-stride over 524288 float4s (2 iterations/thread).
    const int n4 = out_size / 4;
    safety_clip_kernel<<<1024, 256, 0, stream>>>(x, ws, out, n4);
}